// STRNNCell_25718264169356
// MI455X (gfx1250) — compile-verified
//
#include <hip/hip_runtime.h>
#include <hip/hip_bf16.h>
#include <math.h>

// MI455X / gfx1250, wave32. f32 WMMA 16x16x4 for W-streaming matvec strips.
typedef float v2f __attribute__((ext_vector_type(2)));
typedef float v8f __attribute__((ext_vector_type(8)));

#define HH 512
#define LL 256
#define NCHUNK 8            // L-chunks in stage 1

// ---------------------------------------------------------------------------
// Stage 1: fused gather + 4 weighted column reductions, split over L-chunks.
//   rp[c][j][k] = sum_{l in chunk c} w_j[l] * loc_table[loc[l]][k]
// Grid: NCHUNK blocks x 256 threads; thread -> float2 column pair (coalesced).
// ---------------------------------------------------------------------------
__global__ void strnn_gather_partial(const float* __restrict__ td_u,
                                     const float* __restrict__ td_l,
                                     const float* __restrict__ ld_u,
                                     const float* __restrict__ ld_l,
                                     const float* __restrict__ loc_table,
                                     const int*   __restrict__ loc,
                                     float* __restrict__ rp /* NCHUNK*4*HH */) {
  __shared__ float w1[32], w2[32], w3[32], w4[32];
  __shared__ int   off[32];
  const int tid = threadIdx.x;
  const int c   = blockIdx.x;                 // l-chunk
  if (tid < 32) {
    const int l = c * 32 + tid;
    float a = td_u[l], b = td_l[l];
    float it = 1.0f / (a + b);
    float ap = a * it, bp = b * it;
    float cu = ld_u[l], d = ld_l[l];
    float is = 1.0f / (cu + d);
    float al = cu * is, be = d * is;
    w1[tid] = al * ap; w2[tid] = al * bp;
    w3[tid] = be * ap; w4[tid] = be * bp;
    off[tid] = loc[l] * HH;                   // max ~5.7e8, fits in int31
  }
  __syncthreads();
  const int k2 = tid * 2;                     // column pair
  float2 a1 = {0.f, 0.f}, a2 = {0.f, 0.f}, a3 = {0.f, 0.f}, a4 = {0.f, 0.f};
#pragma unroll 8
  for (int l = 0; l < 32; ++l) {
    float2 e = *(const float2*)(loc_table + off[l] + k2);  // 8B aligned
    a1.x = fmaf(w1[l], e.x, a1.x); a1.y = fmaf(w1[l], e.y, a1.y);
    a2.x = fmaf(w2[l], e.x, a2.x); a2.y = fmaf(w2[l], e.y, a2.y);
    a3.x = fmaf(w3[l], e.x, a3.x); a3.y = fmaf(w3[l], e.y, a3.y);
    a4.x = fmaf(w4[l], e.x, a4.x); a4.y = fmaf(w4[l], e.y, a4.y);
  }
  float* base = rp + c * (4 * HH);
  *(float2*)(base + 0 * HH + k2) = a1;
  *(float2*)(base + 1 * HH + k2) = a2;
  *(float2*)(base + 2 * HH + k2) = a3;
  *(float2*)(base + 3 * HH + k2) = a4;
}

// Deterministic fixed-order reduction of the NCHUNK partials -> r[4*HH].
__global__ void strnn_reduce_r(const float* __restrict__ rp,
                               float* __restrict__ r) {
  const int i = blockIdx.x * blockDim.x + threadIdx.x;   // 0..2047
  float s = 0.f;
#pragma unroll
  for (int c = 0; c < NCHUNK; ++c) s += rp[c * (4 * HH) + i];
  r[i] = s;
}

// ---------------------------------------------------------------------------
// WMMA f32 16x16x4 over a K-range of one 16-row strip.
//   A (16x4): lane = M, VGPRs = K{2*hi, 2*hi+1} -> one float2 load per step.
//   B (4x16): lane = N; Bseg already offset by the lane's (padded) column, so
//             every lane does an unconditional LDS read -> no EXEC toggling.
// ---------------------------------------------------------------------------
__device__ __forceinline__ void wmma_mv_range(const float* __restrict__ W, int row,
                                              const float* Bseg, int stride,
                                              int hi, int kbeg, int kend,
                                              v8f& acc) {
  const float* wr = W + row * HH;
#pragma unroll 8
  for (int k0 = kbeg; k0 < kend; k0 += 4) {
    const int ka = k0 + 2 * hi;
    float2 av = *(const float2*)(wr + ka);        // 8B aligned (ka even)
    v2f a; a[0] = av.x; a[1] = av.y;
    v2f b; b[0] = Bseg[ka * stride]; b[1] = Bseg[(ka + 1) * stride];
    acc = __builtin_amdgcn_wmma_f32_16x16x4_f32(false, a, false, b, (short)0,
                                                acc, false, false);
  }
}

// ---------------------------------------------------------------------------
// Stage 2: [p|q](512x2) = [W_th_up | W_th_lo](512x1024) @ [r1;r2 | r3;r4]
// 32 blocks x 128 threads: 4 waves per strip, one per {matrix} x {K half};
// per-wave partial accumulators tree-summed via LDS.
// ---------------------------------------------------------------------------
__global__ void strnn_stage2(const float* __restrict__ Wthu,
                             const float* __restrict__ Wthl,
                             const float* __restrict__ r,
                             float* __restrict__ pq) {
  __shared__ float Bs[2 * HH * 3];              // [k 0..1023][col0,col1,zero-pad]
  __shared__ float red[4][32][8];
  const int tid = threadIdx.x;
  for (int i = tid; i < 2 * HH; i += 128) {
    Bs[3 * i + 0] = r[i];                       // col0 = r1 || r2
    Bs[3 * i + 1] = r[2 * HH + i];              // col1 = r3 || r4
    Bs[3 * i + 2] = 0.f;                        // pad column (lanes n>=2)
  }
  __syncthreads();
  const int lane = tid & 31, w = tid >> 5;
  const int ln = lane & 15, hi = lane >> 4;
  const int m  = blockIdx.x * 16 + ln;
  const int kb = (ln < 2) ? ln : 2;
  const int seg = w >> 1, kh = w & 1;
  const float* W    = seg ? Wthl : Wthu;
  const float* Bseg = Bs + seg * (HH * 3) + kb;
  v8f acc = {0.f, 0.f, 0.f, 0.f, 0.f, 0.f, 0.f, 0.f};
  wmma_mv_range(W, m, Bseg, 3, hi, kh * 256, kh * 256 + 256, acc);
#pragma unroll
  for (int i = 0; i < 8; ++i) red[w][lane][i] = acc[i];
  __syncthreads();
  if (w == 0 && ln < 2) {                       // D: lane=N, M = i + 8*hi
    float* dst = pq + ln * HH + blockIdx.x * 16 + 8 * hi;
#pragma unroll
    for (int i = 0; i < 8; ++i)
      dst[i] = (red[0][lane][i] + red[1][lane][i]) +
               (red[2][lane][i] + red[3][lane][i]);
  }
}

// ---------------------------------------------------------------------------
// Stage 3: out = sigmoid([W_sh_up|W_sh_lo|W_ih](512x1536) @ [p;q;hx])
// 32 blocks x 192 threads: 6 waves, one per {matrix} x {K half}.
// ---------------------------------------------------------------------------
__global__ void strnn_stage3(const float* __restrict__ Wshu,
                             const float* __restrict__ Wshl,
                             const float* __restrict__ Wih,
                             const float* __restrict__ pq,
                             const float* __restrict__ hx,
                             float* __restrict__ out) {
  __shared__ float Bs[3 * HH * 2];              // [k 0..1535][col0, zero-pad]
  __shared__ float red[6][32][8];
  const int tid = threadIdx.x;
  for (int i = tid; i < 3 * HH; i += 192) {
    Bs[2 * i + 0] = (i < 2 * HH) ? pq[i] : hx[i - 2 * HH];  // p || q || hx
    Bs[2 * i + 1] = 0.f;
  }
  __syncthreads();
  const int lane = tid & 31, w = tid >> 5;
  const int ln = lane & 15, hi = lane >> 4;
  const int m  = blockIdx.x * 16 + ln;
  const int kb = (ln == 0) ? 0 : 1;
  const int seg = w >> 1, kh = w & 1;
  const float* W    = (seg == 0) ? Wshu : ((seg == 1) ? Wshl : Wih);
  const float* Bseg = Bs + seg * (HH * 2) + kb;
  v8f acc = {0.f, 0.f, 0.f, 0.f, 0.f, 0.f, 0.f, 0.f};
  wmma_mv_range(W, m, Bseg, 2, hi, kh * 256, kh * 256 + 256, acc);
#pragma unroll
  for (int i = 0; i < 8; ++i) red[w][lane][i] = acc[i];
  __syncthreads();
  if (w == 0 && ln == 0) {
    float* dst = out + blockIdx.x * 16 + 8 * hi;
#pragma unroll
    for (int i = 0; i < 8; ++i) {
      float x = ((red[0][lane][i] + red[1][lane][i]) +
                 (red[2][lane][i] + red[3][lane][i])) +
                (red[4][lane][i] + red[5][lane][i]);
      dst[i] = 1.0f / (1.0f + expf(-x));
    }
  }
}

extern "C" void kernel_launch(void* const* d_in, const int* in_sizes, int n_in,
                              void* d_out, int out_size, void* d_ws, size_t ws_size,
                              hipStream_t stream) {
  const float* td_u  = (const float*)d_in[0];
  const float* td_l  = (const float*)d_in[1];
  const float* ld_u  = (const float*)d_in[2];
  const float* ld_l  = (const float*)d_in[3];
  const float* hx    = (const float*)d_in[4];
  const float* W_ih  = (const float*)d_in[5];
  const float* W_thu = (const float*)d_in[6];
  const float* W_thl = (const float*)d_in[7];
  const float* W_shu = (const float*)d_in[8];
  const float* W_shl = (const float*)d_in[9];
  const float* loc_table = (const float*)d_in[10];
  const int*   loc   = (const int*)d_in[11];
  float* out = (float*)d_out;
  float* ws  = (float*)d_ws;
  float* rp  = ws;                              // NCHUNK * 4*HH
  float* r   = rp + NCHUNK * 4 * HH;            // 4*HH
  float* pq  = r + 4 * HH;                      // 2*HH

  strnn_gather_partial<<<NCHUNK, 256, 0, stream>>>(td_u, td_l, ld_u, ld_l,
                                                   loc_table, loc, rp);
  strnn_reduce_r<<<8, 256, 0, stream>>>(rp, r);
  strnn_stage2<<<HH / 16, 128, 0, stream>>>(W_thu, W_thl, r, pq);
  strnn_stage3<<<HH / 16, 192, 0, stream>>>(W_shu, W_shl, W_ih, pq, hx, out);
}